// TripletLoss_88742614270246
// MI455X (gfx1250) — compile-verified
//
#include <hip/hip_runtime.h>

#define NUM_I 8192
#define NUM_T 8192
#define MARGIN 0.2f

typedef __attribute__((ext_vector_type(4))) float  f4;
typedef __attribute__((ext_vector_type(2))) float  v2f;
typedef __attribute__((ext_vector_type(8))) float  v8f;

// ---------------------------------------------------------------------------
// Kernel 1: one block (256 threads, 8 wave32) per row. Row is fully
// register-resident: each thread holds 32 sim values (8 x float4) plus the
// 32 target flags (8 x packed uchar4). Single pass over HBM (bandwidth bound).
// ---------------------------------------------------------------------------
__global__ __launch_bounds__(256) void
row_hinge_kernel(const float* __restrict__ sim,
                 const unsigned char* __restrict__ tgt,
                 float* __restrict__ per_row)
{
    const int row = blockIdx.x;
    const int tid = threadIdx.x;                       // 0..255
    const size_t base = (size_t)row * NUM_T;

    // ---- load 32 elements / thread, fully coalesced, non-temporal ----
    f4           s[8];
    unsigned int m[8];
    #pragma unroll
    for (int j = 0; j < 8; ++j) {
        const size_t col = (size_t)j * 1024 + (size_t)tid * 4;
        s[j] = __builtin_nontemporal_load((const f4*)(sim + base + col));
        m[j] = __builtin_nontemporal_load((const unsigned int*)(tgt + base + col));
    }

    // ---- local masked min over positives ----
    float vmin = 1e30f;
    #pragma unroll
    for (int j = 0; j < 8; ++j) {
        #pragma unroll
        for (int c = 0; c < 4; ++c) {
            const bool pos = ((m[j] >> (8 * c)) & 0xFFu) != 0u;
            if (pos) vmin = fminf(vmin, s[j][c]);
        }
    }

    // ---- block min reduction (deterministic LDS tree) ----
    __shared__ float red[256];
    red[tid] = vmin;
    __syncthreads();
    #pragma unroll
    for (int st = 128; st > 0; st >>= 1) {
        if (tid < st) red[tid] = fminf(red[tid], red[tid + st]);
        __syncthreads();
    }
    const float row_min = red[0];
    __syncthreads();                                   // everyone read red[0]

    // ---- hinge sum from registers (no re-read of HBM) ----
    float acc = 0.0f;
    #pragma unroll
    for (int j = 0; j < 8; ++j) {
        #pragma unroll
        for (int c = 0; c < 4; ++c) {
            const bool pos = ((m[j] >> (8 * c)) & 0xFFu) != 0u;
            const float h = fmaxf(MARGIN + s[j][c] - row_min, 0.0f);
            acc += pos ? 0.0f : h;
        }
    }

    // ---- block sum reduction ----
    red[tid] = acc;
    __syncthreads();
    #pragma unroll
    for (int st = 128; st > 0; st >>= 1) {
        if (tid < st) red[tid] += red[tid + st];
        __syncthreads();
    }
    if (tid == 0) per_row[row] = red[0];
}

// ---------------------------------------------------------------------------
// Kernel 2: mean of 8192 per-row sums via V_WMMA_F32_16X16X4_F32 with an
// all-ones B matrix. D[m,n] += sum_k A[m,k]; since B is all ones the A lane
// layout is irrelevant for a total sum: each lane streams 2 consecutive
// floats, and the sum of all D entries equals 16x the true total.
// One block = 8 waves, each wave accumulates 16 WMMA chunks of 64 values.
// ---------------------------------------------------------------------------
__global__ __launch_bounds__(256) void
final_mean_kernel(const float* __restrict__ per_row, float* __restrict__ out)
{
    const int tid  = threadIdx.x;
    const int wave = tid >> 5;
    const int lane = tid & 31;

    v8f acc = {};
    v2f bones;
    bones[0] = 1.0f;
    bones[1] = 1.0f;

    #pragma unroll
    for (int i = 0; i < 16; ++i) {
        const int chunk = wave + 8 * i;                // 0..127, 64 floats each
        const float* p = per_row + chunk * 64 + lane * 2;
        v2f a;
        a[0] = p[0];
        a[1] = p[1];
        acc = __builtin_amdgcn_wmma_f32_16x16x4_f32(
            /*neg_a=*/false, a, /*neg_b=*/false, bones,
            /*c_mod=*/(short)0, acc, /*reuse_a=*/false, /*reuse_b=*/false);
    }

    // per-lane sum of the 8 accumulator VGPRs
    float s = acc[0] + acc[1] + acc[2] + acc[3] + acc[4] + acc[5] + acc[6] + acc[7];

    __shared__ float red[256];
    red[tid] = s;
    __syncthreads();
    #pragma unroll
    for (int st = 128; st > 0; st >>= 1) {
        if (tid < st) red[tid] += red[tid + st];
        __syncthreads();
    }
    if (tid == 0)
        out[0] = red[0] * (1.0f / (16.0f * (float)NUM_I));  // /16 (col replication) /8192 (mean)
}

// ---------------------------------------------------------------------------
extern "C" void kernel_launch(void* const* d_in, const int* in_sizes, int n_in,
                              void* d_out, int out_size, void* d_ws, size_t ws_size,
                              hipStream_t stream)
{
    const float*         sim = (const float*)d_in[0];
    const unsigned char* tgt = (const unsigned char*)d_in[1]; // numpy bool = 1 byte
    float*               per_row = (float*)d_ws;              // 8192 floats scratch
    float*               out = (float*)d_out;

    row_hinge_kernel<<<dim3(NUM_I), dim3(256), 0, stream>>>(sim, tgt, per_row);
    final_mean_kernel<<<dim3(1), dim3(256), 0, stream>>>(per_row, out);
}